// CrossMarketCompoundEmbedding_3478923510364
// MI455X (gfx1250) — compile-verified
//
#include <hip/hip_runtime.h>
#include <cstdint>

// Reference constants
#define N_ROWS      65536
#define EMBED_DIM   512
#define SEG4        32            // float4s per 128-float segment
#define ROWCLASSES  12            // lcm(3,4): distinct row patterns
#define PAT4        (ROWCLASSES * (EMBED_DIM / 4))   // 1536 float4 = 24 KB
#define THREADS     256
#define BLOCKS      1024
#define TOTAL_T     (THREADS * BLOCKS)               // 262144 = 2^18
#define TOTAL4      (N_ROWS * (EMBED_DIM / 4))       // 8388608 float4
#define ITERS       (TOTAL4 / TOTAL_T)               // exactly 32

__global__ __launch_bounds__(THREADS)
void CrossMarketCompoundEmbedding_kernel(const float4* __restrict__ price4,
                                         const float4* __restrict__ size4,
                                         const float4* __restrict__ exch4,
                                         const float4* __restrict__ pair4,
                                         float4* __restrict__ out4)
{
    __shared__ float4 pat[PAT4];   // the 12 unique output rows (24 KB)

    const int tid = threadIdx.x;

    // ---- Stage the 12-row pattern into LDS (6 float4 per thread) ----
    #pragma unroll
    for (int k = 0; k < PAT4 / THREADS; ++k) {
        int s   = tid + k * THREADS;      // pattern float4 index 0..1535
        int rc  = s >> 7;                 // row class 0..11
        int c4  = s & 127;                // float4 column within row
        float4 v;
        if (c4 < SEG4)            v = price4[c4];
        else if (c4 < 2 * SEG4)   v = size4[c4 - SEG4];
        else if (c4 < 3 * SEG4)   v = exch4[(rc % 3) * SEG4 + (c4 - 2 * SEG4)];
        else                      v = pair4[(rc & 3) * SEG4 + (c4 - 3 * SEG4)];
        pat[s] = v;
    }
    __syncthreads();

    // ---- Stream pattern to global with async LDS->global B128 stores ----
    const uint32_t t = blockIdx.x * THREADS + tid;

    // LDS byte offset of this thread's first source float4.
    // Generic LDS pointer: low 32 bits == LDS byte address (shared aperture).
    const uint32_t ldsBase  = (uint32_t)(uintptr_t)(&pat[0]);
    uint32_t m              = (t % PAT4) * 16u;                // 0..24560
    const uint32_t stepMod  = (uint32_t)((TOTAL_T % PAT4) * 16u); // 16384
    const uint32_t patBytes = PAT4 * 16u;                      // 24576

    float4* gp = out4 + t;   // advances by TOTAL_T float4s (4 MB) per iter

    #pragma unroll 4
    for (int j = 0; j < ITERS; ++j) {
        uint32_t lds_addr = ldsBase + m;
        // vaddr (64-bit global address), vdata (32-bit LDS byte address)
        asm volatile("global_store_async_from_lds_b128 %0, %1, off"
                     :
                     : "v"(gp), "v"(lds_addr)
                     : "memory");
        gp += TOTAL_T;
        m += stepMod;
        if (m >= patBytes) m -= patBytes;
    }

    // Drain the async queue (S_ENDPGM also waits idle; this is explicit).
    asm volatile("s_wait_asynccnt 0" ::: "memory");
}

extern "C" void kernel_launch(void* const* d_in, const int* in_sizes, int n_in,
                              void* d_out, int out_size, void* d_ws, size_t ws_size,
                              hipStream_t stream)
{
    // d_in order: [0]=num_features(int,1), [1]=price_w(128 f32),
    // [2]=size_w(128 f32), [3]=exchange_w(512 f32), [4]=pair_w(640 f32)
    const float4* price4 = (const float4*)d_in[1];
    const float4* size4  = (const float4*)d_in[2];
    const float4* exch4  = (const float4*)d_in[3];
    const float4* pair4  = (const float4*)d_in[4];
    float4* out4 = (float4*)d_out;

    CrossMarketCompoundEmbedding_kernel<<<BLOCKS, THREADS, 0, stream>>>(
        price4, size4, exch4, pair4, out4);
}